// CTCLayer_15942918603183
// MI455X (gfx1250) — compile-verified
//
#include <hip/hip_runtime.h>
#include <hip/hip_bf16.h>

// CTC batch cost for MI455X (gfx1250).
// Pass 1: row-sum denominators via V_WMMA_F32_16X16X4_F32 against a ones matrix
//         (full f32 precision, streams all of y_pred once -> HBM-bound ~4.3us).
// Pass 2: alpha recursion, one wave32 per batch element, lane-local 5-state
//         blocks + __shfl_up for neighbor states, gathers hit L2 (y_pred fits
//         in the 192MB L2 after pass 1).

#define CTC_B 1024
#define CTC_T 256
#define CTC_C 96
#define CTC_L 64
#define CTC_S 129            // 2L+1
#define NEGV  (-1e30f)
#define EPSV  (1e-7f)

typedef float v2f __attribute__((ext_vector_type(2)));
typedef float v8f __attribute__((ext_vector_type(8)));

// ---------------------------------------------------------------------------
// Pass 1: denom[b,t] = log(sum_c y_pred[b,t,c] + C*eps)
// One wave per 16-timestep tile. A = [16 x 4] f32 chunk of y_pred rows,
// B = ones[4 x 16], accumulate 24 chunks -> D[m,n] = rowsum(m) for every n.
// ---------------------------------------------------------------------------
__global__ __launch_bounds__(256) void ctc_denom_wmma(
    const float* __restrict__ y_pred, float* __restrict__ denom) {
  const int wave = (int)((blockIdx.x * blockDim.x + threadIdx.x) >> 5);
  const int lane = (int)(threadIdx.x & 31);

  const int b  = wave >> 4;          // 16 tiles of 16 timesteps per batch row
  const int t0 = (wave & 15) << 4;
  const int m  = lane & 15;          // tile row handled by this lane
  const int kp = (lane >> 4) << 1;   // K pair: lanes 0-15 -> K=0,1 ; 16-31 -> K=2,3

  const float* rowp = y_pred + (size_t)(b * CTC_T + t0 + m) * CTC_C + kp;

  v8f acc = {};
  v2f bones;  bones.x = 1.0f;  bones.y = 1.0f;

#pragma unroll
  for (int kc = 0; kc < CTC_C / 4; ++kc) {
    v2f a = *(const v2f*)(rowp + kc * 4);      // aligned float2 (even offsets)
    acc = __builtin_amdgcn_wmma_f32_16x16x4_f32(
        /*neg_a=*/false, a, /*neg_b=*/false, bones,
        /*c_mod=*/(short)0, acc, /*reuse_a=*/false, /*reuse_b=*/false);
  }

  // D layout: lanes 0-15 VGPR r = D[M=r, N=lane]; lanes 16-31 VGPR r = D[M=8+r].
  const int base = b * CTC_T + t0;
  if (lane == 0) {
#pragma unroll
    for (int r = 0; r < 8; ++r)
      denom[base + r] = __logf(acc[r] + (float)CTC_C * EPSV);
  } else if (lane == 16) {
#pragma unroll
    for (int r = 0; r < 8; ++r)
      denom[base + 8 + r] = __logf(acc[r] + (float)CTC_C * EPSV);
  }
}

// ---------------------------------------------------------------------------
// Pass 2: forward (alpha) recursion. One wave32 per batch element.
// Lane l owns states s = 5l .. 5l+4 (s < 129). Even states are blank.
// Per timestep the wave stages the 65 needed log-probs (64 labels + blank)
// into LDS, then each lane updates its 5 states; neighbor alphas via shfl_up.
// ---------------------------------------------------------------------------
__global__ __launch_bounds__(32) void ctc_alpha_scan(
    const float* __restrict__ y_pred, const int* __restrict__ y_true,
    const float* __restrict__ denom, float* __restrict__ out) {
  const int b    = (int)blockIdx.x;
  const int lane = (int)threadIdx.x;

  __shared__ int   labs[CTC_L];
  __shared__ float lp[CTC_L + 1];          // [0..63]=labels, [64]=blank

  labs[lane]      = y_true[b * CTC_L + lane];
  labs[lane + 32] = y_true[b * CTC_L + lane + 32];
  __syncthreads();

  // Per-state metadata (5 states per lane).
  const int s0 = lane * 5;
  bool valid[5];
  bool skipf[5];
  int  lpidx[5];
#pragma unroll
  for (int j = 0; j < 5; ++j) {
    const int  s   = s0 + j;
    const bool od  = (s & 1) != 0;
    const int  lab = (s - 1) >> 1;
    valid[j] = (s < CTC_S);
    lpidx[j] = (od && lab < CTC_L) ? lab : CTC_L;   // clamped, blank slot = L
    bool sk = false;
    if (od && lab >= 1 && lab < CTC_L) sk = (labs[lab] != labs[lab - 1]);
    skipf[j] = sk;
  }

  float av[5];

  // ---- t = 0: alpha0[0] = lp(blank), alpha0[1] = lp(label0), else NEG ----
  {
    const float  dl  = denom[b * CTC_T + 0];
    const float* row = y_pred + (size_t)(b * CTC_T + 0) * CTC_C;
    lp[lane]      = __logf(row[labs[lane]]      + EPSV) - dl;
    lp[lane + 32] = __logf(row[labs[lane + 32]] + EPSV) - dl;
    if (lane == 0) lp[CTC_L] = __logf(row[CTC_C - 1] + EPSV) - dl;
    __syncthreads();
#pragma unroll
    for (int j = 0; j < 5; ++j) {
      const int s = s0 + j;
      av[j] = (valid[j] && s <= 1) ? lp[lpidx[j]] : NEGV;
    }
    __syncthreads();
  }

  // ---- t = 1 .. T-1 ----
  for (int t = 1; t < CTC_T; ++t) {
    const float  dl  = denom[b * CTC_T + t];
    const float* row = y_pred + (size_t)(b * CTC_T + t) * CTC_C;
    lp[lane]      = __logf(row[labs[lane]]      + EPSV) - dl;
    lp[lane + 32] = __logf(row[labs[lane + 32]] + EPSV) - dl;
    if (lane == 0) lp[CTC_L] = __logf(row[CTC_C - 1] + EPSV) - dl;
    __syncthreads();

    float pa3 = __shfl_up(av[3], 1);   // prev lane's s-2 for my state 0
    float pa4 = __shfl_up(av[4], 1);   // prev lane's s-1 for my state 0
    if (lane == 0) { pa3 = NEGV; pa4 = NEGV; }   // s=0 has no predecessors

    float nv[5];
#pragma unroll
    for (int j = 0; j < 5; ++j) {
      const float x0 = av[j];
      const float x1 = (j == 0) ? pa4 : av[j - 1];
      float x2 = (j == 0) ? pa3 : ((j == 1) ? pa4 : av[j - 2]);
      x2 = skipf[j] ? x2 : NEGV;
      const float mx  = fmaxf(x0, fmaxf(x1, x2));
      const float lse = mx + __logf(__expf(x0 - mx) + __expf(x1 - mx) +
                                    __expf(x2 - mx));
      nv[j] = valid[j] ? (lse + lp[lpidx[j]]) : NEGV;
    }
#pragma unroll
    for (int j = 0; j < 5; ++j) av[j] = nv[j];
    __syncthreads();
  }

  // ---- loss = -logsumexp(alpha[S-1], alpha[S-2]); both live on lane 25 ----
  if (lane == 25) {
    const float x0 = av[3];   // s = 128
    const float x1 = av[2];   // s = 127
    const float mx = fmaxf(x0, x1);
    out[b] = -(mx + __logf(__expf(x0 - mx) + __expf(x1 - mx)));
  }
}

// ---------------------------------------------------------------------------
extern "C" void kernel_launch(void* const* d_in, const int* in_sizes, int n_in,
                              void* d_out, int out_size, void* d_ws, size_t ws_size,
                              hipStream_t stream) {
  (void)in_sizes; (void)n_in; (void)out_size; (void)ws_size;
  const int*   y_true = (const int*)d_in[0];     // [B, L] labels
  const float* y_pred = (const float*)d_in[1];   // [B, T, C]
  float* denom = (float*)d_ws;                   // [B*T] scratch (1 MB)
  float* out   = (float*)d_out;                  // [B, 1]

  // Pass 1: B*T/16 = 16384 waves, 8 waves per 256-thread block -> 2048 blocks.
  ctc_denom_wmma<<<dim3((CTC_B * CTC_T / 16) / 8), dim3(256), 0, stream>>>(
      y_pred, denom);
  // Pass 2: one single-wave workgroup per batch element.
  ctc_alpha_scan<<<dim3(CTC_B), dim3(32), 0, stream>>>(
      y_pred, y_true, denom, out);
}